// NormalGCN_layer_79199196938454
// MI455X (gfx1250) — compile-verified
//
#include <hip/hip_runtime.h>
#include <hip/hip_bf16.h>

#define NN 100000
#define EE 1600000
#define DD 128

typedef __attribute__((ext_vector_type(2))) float v2f;
typedef __attribute__((ext_vector_type(8))) float v8f;

// ---------------------------------------------------------------------------
// 1) Transpose both weight matrices once: Wt[k*128+j] = W[j*128+k]
//    (makes WMMA B-fragment loads coalesced: a row of Wt is a column of W)
// ---------------------------------------------------------------------------
__global__ __launch_bounds__(256) void transpose_kernel(
    const float* __restrict__ A_w, const float* __restrict__ B_w,
    float* __restrict__ WtA, float* __restrict__ WtB)
{
    __shared__ float tile[16][17];
    const float* Win = blockIdx.z ? B_w : A_w;
    float* Wout      = blockIdx.z ? WtB : WtA;
    const int j0 = blockIdx.x * 16, k0 = blockIdx.y * 16;
    const int tx = threadIdx.x & 15, ty = threadIdx.x >> 4;
    tile[ty][tx] = Win[(j0 + ty) * DD + k0 + tx];
    __syncthreads();
    Wout[(k0 + ty) * DD + j0 + tx] = tile[tx][ty];
}

// ---------------------------------------------------------------------------
// 2) Zero-init indeg[N] and BN partial sums[256]
// ---------------------------------------------------------------------------
__global__ __launch_bounds__(256) void init_kernel(int* __restrict__ indeg,
                                                   float* __restrict__ sums)
{
    const int i = blockIdx.x * 256 + threadIdx.x;
    if (i < NN) indeg[i] = 0;
    if (i < 256) sums[i] = 0.0f;
}

// ---------------------------------------------------------------------------
// 3) Fused dual GEMM via V_WMMA_F32_16X16X4_F32 (exact fp32 path).
//    Block = 256 threads = 8 waves. Wave w: rows [blk*64 + (w&3)*16, +16),
//    matrix = w>>2 (0 -> AX into outA, 1 -> BX into outB).
//    Per wave: 8 accumulator tiles (16x16) covering all 128 output columns.
//    A-frag (16x4 f32): lanes 0-15 K={k,k+1} in v0/v1, lanes 16-31 K={k+2,k+3}.
//    B-frag (4x16 f32): mirrored K striping, N across the 16-lane halves.
// ---------------------------------------------------------------------------
__global__ __launch_bounds__(256) void gemm_kernel(
    const float* __restrict__ X,
    const float* __restrict__ WtA, const float* __restrict__ WtB,
    const float* __restrict__ bA,  const float* __restrict__ bB,
    float* __restrict__ outA, float* __restrict__ outB)
{
    const int wave    = threadIdx.x >> 5;     // 0..7
    const int lane    = threadIdx.x & 31;
    const int mat     = wave >> 2;            // 0 = A, 1 = B
    const int rowTile = wave & 3;
    const int row0    = blockIdx.x * 64 + rowTile * 16;

    const float* Wt   = mat ? WtB : WtA;
    const float* bias = mat ? bB  : bA;
    float* outp       = mat ? outB : outA;

    const int mrow  = lane & 15;
    const int khalf = lane >> 4;              // 0 or 1

    int arow = row0 + mrow;
    if (arow >= NN) arow = NN - 1;            // clamp (EXEC must stay all-1s)
    const float* xrow = X + (size_t)arow * DD + 2 * khalf;

    v8f acc[8] = {};

    for (int k = 0; k < DD; k += 4) {
        v2f a;
        a.x = xrow[k];
        a.y = xrow[k + 1];
        #pragma unroll
        for (int t = 0; t < 8; ++t) {
            const int col = t * 16 + mrow;
            v2f b;
            b.x = Wt[(k + 2 * khalf) * DD + col];
            b.y = Wt[(k + 2 * khalf + 1) * DD + col];
            acc[t] = __builtin_amdgcn_wmma_f32_16x16x4_f32(
                false, a, false, b, (short)0, acc[t], false, false);
        }
    }

    // Epilogue: bias add + store. C/D layout: VGPR r -> M = r + 8*khalf, N = lane&15.
    #pragma unroll
    for (int t = 0; t < 8; ++t) {
        const int col = t * 16 + mrow;
        const float bv = bias[col];
        #pragma unroll
        for (int r = 0; r < 8; ++r) {
            const int row = row0 + r + 8 * khalf;
            if (row < NN) outp[(size_t)row * DD + col] = acc[t][r] + bv;
        }
    }
}

// ---------------------------------------------------------------------------
// 4) Edge scatter: ACC[dst] += BX[src] (ACC was pre-seeded with AX by GEMM).
//    One wave per edge, float4 per lane (128 cols / 32 lanes), hardware fp32
//    atomics; BX/ACC footprints (51 MB each) are L2-resident.
// ---------------------------------------------------------------------------
__global__ __launch_bounds__(256) void scatter_kernel(
    const float* __restrict__ BX,
    const int* __restrict__ src, const int* __restrict__ dst,
    float* __restrict__ ACC, int* __restrict__ indeg)
{
    const int lane = threadIdx.x & 31;
    const int e    = blockIdx.x * 8 + (threadIdx.x >> 5);
    if (e >= EE) return;
    const int s = src[e];
    const int d = dst[e];
    if (lane == 0) atomicAdd(&indeg[d], 1);
    const float4 v = ((const float4*)(BX + (size_t)s * DD))[lane];
    float* drow = ACC + (size_t)d * DD + lane * 4;
    unsafeAtomicAdd(drow + 0, v.x);
    unsafeAtomicAdd(drow + 1, v.y);
    unsafeAtomicAdd(drow + 2, v.z);
    unsafeAtomicAdd(drow + 3, v.w);
}

// ---------------------------------------------------------------------------
// 5) BN batch statistics: per-column sum and sum-of-squares of
//    H = (indeg>0 ? ACC : X) * snorm_n. 128 threads = 128 columns, each block
//    reduces 256 rows locally, then 2 fp atomics per thread.
// ---------------------------------------------------------------------------
__global__ __launch_bounds__(128) void stats_kernel(
    const float* __restrict__ ACC, const float* __restrict__ X,
    const float* __restrict__ snorm, const int* __restrict__ indeg,
    float* __restrict__ sums)
{
    const int d    = threadIdx.x;
    const int row0 = blockIdx.x * 256;
    const int rend = (row0 + 256 < NN) ? row0 + 256 : NN;
    float s = 0.0f, ss = 0.0f;
    for (int r = row0; r < rend; ++r) {
        const size_t idx = (size_t)r * DD + d;
        float h = (indeg[r] > 0) ? ACC[idx] : X[idx];
        h *= snorm[r];
        s  += h;
        ss += h * h;
    }
    unsafeAtomicAdd(&sums[d], s);
    unsafeAtomicAdd(&sums[DD + d], ss);
}

// ---------------------------------------------------------------------------
// 6) Finalize in place: H = X + relu(BN(H * snorm)), written over ACC (d_out).
// ---------------------------------------------------------------------------
__global__ __launch_bounds__(256) void finalize_kernel(
    float* __restrict__ ACC, const float* __restrict__ X,
    const float* __restrict__ snorm, const int* __restrict__ indeg,
    const float* __restrict__ gamma, const float* __restrict__ beta,
    const float* __restrict__ sums)
{
    const size_t i = (size_t)blockIdx.x * 256 + threadIdx.x;
    if (i >= (size_t)NN * DD) return;
    const int r = (int)(i >> 7);
    const int d = (int)(i & (DD - 1));
    const float inv_n = 1.0f / (float)NN;
    const float mean  = sums[d] * inv_n;
    const float var   = sums[DD + d] * inv_n - mean * mean;
    const float rstd  = __frsqrt_rn(var + 1e-5f);
    const float x = X[i];
    float h = (indeg[r] > 0) ? ACC[i] : x;
    h *= snorm[r];
    h = (h - mean) * rstd * gamma[d] + beta[d];
    h = fmaxf(h, 0.0f);
    ACC[i] = x + h;
}

// ---------------------------------------------------------------------------
// Orchestration.
// Inputs: 0:X 1:E_X 2:snorm_n 3:snorm_e 4:src 5:dst 6:A_w 7:A_b 8:B_w 9:B_b
//         10:gamma 11:beta
// d_out = [ H (N*D floats) | E_X (E*D floats) ].
// BX is staged in the E_X tail of d_out and overwritten by the final copy.
// Workspace: WtA(64K) | WtB(64K) | indeg(400K) | sums(1K)  (~0.52 MB)
// ---------------------------------------------------------------------------
extern "C" void kernel_launch(void* const* d_in, const int* in_sizes, int n_in,
                              void* d_out, int out_size, void* d_ws, size_t ws_size,
                              hipStream_t stream)
{
    const float* X     = (const float*)d_in[0];
    const float* E_X   = (const float*)d_in[1];
    const float* snorm = (const float*)d_in[2];
    const int*   src   = (const int*)d_in[4];
    const int*   dst   = (const int*)d_in[5];
    const float* A_w   = (const float*)d_in[6];
    const float* A_b   = (const float*)d_in[7];
    const float* B_w   = (const float*)d_in[8];
    const float* B_b   = (const float*)d_in[9];
    const float* gamma = (const float*)d_in[10];
    const float* beta  = (const float*)d_in[11];

    char* ws = (char*)d_ws;
    float* WtA   = (float*)(ws + 0);
    float* WtB   = (float*)(ws + 65536);
    int*   indeg = (int*)  (ws + 131072);
    float* sums  = (float*)(ws + 531456);

    float* H  = (float*)d_out;                       // N*D accumulator / output
    float* BX = (float*)d_out + (size_t)NN * DD;     // staged in E_X region

    transpose_kernel<<<dim3(8, 8, 2), 256, 0, stream>>>(A_w, B_w, WtA, WtB);
    init_kernel<<<(NN + 255) / 256, 256, 0, stream>>>(indeg, sums);
    gemm_kernel<<<(NN + 63) / 64, 256, 0, stream>>>(X, WtA, WtB, A_b, B_b, H, BX);
    scatter_kernel<<<EE / 8, 256, 0, stream>>>(BX, src, dst, H, indeg);
    stats_kernel<<<(NN + 255) / 256, 128, 0, stream>>>(H, X, snorm, indeg, sums);
    finalize_kernel<<<(int)(((size_t)NN * DD + 255) / 256), 256, 0, stream>>>(
        H, X, snorm, indeg, gamma, beta, sums);
    hipMemcpyAsync((float*)d_out + (size_t)NN * DD, E_X,
                   (size_t)EE * DD * sizeof(float), hipMemcpyDeviceToDevice, stream);
}